// LinOSSLayer_28707561406821
// MI455X (gfx1250) — compile-verified
//
#include <hip/hip_runtime.h>
#include <hip/hip_bf16.h>
#include <math.h>

typedef __attribute__((ext_vector_type(2))) float v2f;
typedef __attribute__((ext_vector_type(8))) float v8f;

#define L_SEQ 16384
#define H_DIM 128
#define P_DIM 256
#define N2    512           // 2*P (re||im)
#define NCHUNK 128
#define TCHUNK 128          // NCHUNK*TCHUNK == L_SEQ

// workspace float offsets
#define OFF_M11  ((size_t)0)
#define OFF_M12  ((size_t)256)
#define OFF_M21  ((size_t)512)
#define OFF_M22  ((size_t)768)
#define OFF_C1   ((size_t)1024)   // st*M11  (scale applied to Bu -> F1)
#define OFF_RAT  ((size_t)1280)   // M21/M11 (f2 = rat*f1)
#define OFF_W1   ((size_t)2048)                      // W1T: 512(n) x 128(k)
#define OFF_WC   (OFF_W1 + (size_t)N2*H_DIM)         // WcT: 128(h) x 512(k)
#define OFF_S    (OFF_WC + (size_t)H_DIM*N2)         // NCHUNK x 1024
#define OFF_CA   (OFF_S  + (size_t)NCHUNK*1024)      // NCHUNK x 1024
#define OFF_F1   (OFF_CA + (size_t)NCHUNK*1024)      // L x 512 (reused as X2)

// ---------------- per-state parameters ----------------
__global__ void prep_params(const float* __restrict__ A_diag,
                            const float* __restrict__ steps,
                            float* __restrict__ ws) {
    int p = threadIdx.x;
    float A  = fmaxf(A_diag[p], 0.0f);
    float st = 1.0f / (1.0f + expf(-steps[p]));
    float schur = 1.0f / (1.0f + st * st * A);
    float m11 = 1.0f - st * st * A * schur;      // == schur
    float m12 = -st * A * schur;
    float m21 = st * schur;
    float m22 = schur;
    ws[OFF_M11 + p] = m11;
    ws[OFF_M12 + p] = m12;
    ws[OFF_M21 + p] = m21;
    ws[OFF_M22 + p] = m22;
    ws[OFF_C1  + p] = st * m11;
    ws[OFF_RAT + p] = m21 / m11;                 // == st (exact algebra)
}

// W1T[n][k]: n=0..511, k=0..127 (h) : n<256 -> B_re[n][k], else B_im[n-256][k]
__global__ void build_w1(const float* __restrict__ B, float* __restrict__ ws) {
    int idx = blockIdx.x * blockDim.x + threadIdx.x;   // 512*128
    int n = idx >> 7, k = idx & 127;
    float v = (n < P_DIM) ? B[(size_t)n * H_DIM * 2 + k * 2 + 0]
                          : B[(size_t)(n - P_DIM) * H_DIM * 2 + k * 2 + 1];
    ws[OFF_W1 + (size_t)n * H_DIM + k] = v;
}

// WcT[h][k]: h=0..127, k=0..511 : k<256 -> C_re[h][k], else -C_im[h][k-256]
__global__ void build_wc(const float* __restrict__ C, float* __restrict__ ws) {
    int idx = blockIdx.x * blockDim.x + threadIdx.x;   // 128*512
    int h = idx >> 9, k = idx & 511;
    float v = (k < P_DIM) ? C[(size_t)h * P_DIM * 2 + k * 2 + 0]
                          : -C[(size_t)h * P_DIM * 2 + (k - P_DIM) * 2 + 1];
    ws[OFF_WC + (size_t)h * N2 + k] = v;
}

// ---------------- GEMM1: F1 = (st*M11) * (u @ W1)  (L x 512) ----------------
// block: 256 thr (8 waves, 4M x 2N), tile 64(M) x 64(N), K=128 in 2 LDS chunks
// Both A and B tiles K-contiguous in LDS -> b64 fragment loads, no repack.
__global__ void gemm1(const float* __restrict__ u, const float* __restrict__ w1t,
                      const float* __restrict__ ws_par, float* __restrict__ F1) {
    __shared__ float As[64 * 68];    // [m][k]
    __shared__ float Bs[64 * 68];    // [n][k]
    const int tid  = threadIdx.x;
    const int lane = tid & 31;
    const int wave = tid >> 5;
    const int wm = wave & 3, wn = wave >> 2;
    const int r    = lane & 15;
    const int koff = (lane >> 4) << 1;
    const int rowBase = blockIdx.x * 64;
    const int nBase   = blockIdx.y * 64;
    const int r0 = tid >> 2, q = tid & 3;

    v8f acc[2] = {};
    for (int kc = 0; kc < H_DIM; kc += 64) {
        #pragma unroll
        for (int i = 0; i < 4; ++i) {
            int c4 = q + i * 4;
            float4 v = *(const float4*)(u + (size_t)(rowBase + r0) * H_DIM + kc + c4 * 4);
            *(float4*)(As + r0 * 68 + c4 * 4) = v;
        }
        #pragma unroll
        for (int i = 0; i < 4; ++i) {
            int c4 = q + i * 4;
            float4 v = *(const float4*)(w1t + (size_t)(nBase + r0) * H_DIM + kc + c4 * 4);
            *(float4*)(Bs + r0 * 68 + c4 * 4) = v;
        }
        __syncthreads();
        #pragma unroll
        for (int k = 0; k < 64; k += 4) {
            int kk = k + koff;
            v2f a = *(const v2f*)(As + (wm * 16 + r) * 68 + kk);
            #pragma unroll
            for (int nt = 0; nt < 2; ++nt) {
                int col = wn * 32 + nt * 16 + r;
                v2f b = *(const v2f*)(Bs + col * 68 + kk);
                acc[nt] = __builtin_amdgcn_wmma_f32_16x16x4_f32(
                    false, a, false, b, (short)0, acc[nt], false, false);
            }
        }
        __syncthreads();
    }
    const int mhi = (lane >> 4) * 8;
    #pragma unroll
    for (int nt = 0; nt < 2; ++nt) {
        int col = nBase + wn * 32 + nt * 16 + r;
        float c1 = ws_par[OFF_C1 + (col & 255)];
        #pragma unroll
        for (int e = 0; e < 8; ++e) {
            int row = rowBase + wm * 16 + e + mhi;
            F1[(size_t)row * N2 + col] = c1 * acc[nt][e];
        }
    }
}

// ---------------- scan phase A: per-chunk end states (zero init) ----------------
__global__ void scan_partial(const float* __restrict__ F1, const float* __restrict__ ws_par,
                             float* __restrict__ S) {
    int c = threadIdx.x;         // 0..511
    int ch = blockIdx.x;         // 0..127
    int p = c & 255;
    float m11 = ws_par[OFF_M11 + p], m12 = ws_par[OFF_M12 + p];
    float m21 = ws_par[OFF_M21 + p], m22 = ws_par[OFF_M22 + p];
    float rat = ws_par[OFF_RAT + p];
    float x1 = 0.0f, x2 = 0.0f;
    size_t base = (size_t)ch * TCHUNK * N2 + c;
    for (int j = 0; j < TCHUNK; ++j) {
        float f1 = F1[base + (size_t)j * N2];
        float f2 = rat * f1;
        float n1 = m11 * x1 + m12 * x2 + f1;
        float n2 = m21 * x1 + m22 * x2 + f2;
        x1 = n1; x2 = n2;
    }
    S[(size_t)ch * 1024 + c]       = x1;
    S[(size_t)ch * 1024 + 512 + c] = x2;
}

// ---------------- scan phase B: carry prefix with M^TCHUNK ----------------
__global__ void scan_carry(const float* __restrict__ ws_par, const float* __restrict__ S,
                           float* __restrict__ Ca) {
    int c = threadIdx.x;         // 0..511
    int p = c & 255;
    float a = ws_par[OFF_M11 + p], b = ws_par[OFF_M12 + p];
    float g = ws_par[OFF_M21 + p], d = ws_par[OFF_M22 + p];
    // M^128 by 7 squarings
    #pragma unroll
    for (int i = 0; i < 7; ++i) {
        float na = a * a + b * g, nb = a * b + b * d;
        float ng = g * a + d * g, nd = g * b + d * d;
        a = na; b = nb; g = ng; d = nd;
    }
    float ca1 = 0.0f, ca2 = 0.0f;
    for (int ch = 0; ch < NCHUNK; ++ch) {
        Ca[(size_t)ch * 1024 + c]       = ca1;
        Ca[(size_t)ch * 1024 + 512 + c] = ca2;
        float s1 = S[(size_t)ch * 1024 + c];
        float s2 = S[(size_t)ch * 1024 + 512 + c];
        float n1 = a * ca1 + b * ca2 + s1;
        float n2 = g * ca1 + d * ca2 + s2;
        ca1 = n1; ca2 = n2;
    }
}

// ---------------- scan phase C: final scan, write x2 in place over F1 ----------------
__global__ void scan_final(float* __restrict__ F1, const float* __restrict__ ws_par,
                           const float* __restrict__ Ca) {
    int c = threadIdx.x;
    int ch = blockIdx.x;
    int p = c & 255;
    float m11 = ws_par[OFF_M11 + p], m12 = ws_par[OFF_M12 + p];
    float m21 = ws_par[OFF_M21 + p], m22 = ws_par[OFF_M22 + p];
    float rat = ws_par[OFF_RAT + p];
    float x1 = Ca[(size_t)ch * 1024 + c];
    float x2 = Ca[(size_t)ch * 1024 + 512 + c];
    size_t base = (size_t)ch * TCHUNK * N2 + c;
    for (int j = 0; j < TCHUNK; ++j) {
        size_t idx = base + (size_t)j * N2;
        float f1 = F1[idx];
        float f2 = rat * f1;
        float n1 = m11 * x1 + m12 * x2 + f1;
        float n2 = m21 * x1 + m22 * x2 + f2;
        x1 = n1; x2 = n2;
        F1[idx] = x2;               // X2 aliases F1
    }
}

// ---------------- GEMM2: out = X2 @ Wc + u*D  (L x 128) ----------------
// block: 256 thr (8 waves, 4M x 2N), tile 64(M) x 128(N), K=512 in 8 LDS chunks
__global__ void gemm2(const float* __restrict__ X2, const float* __restrict__ wct,
                      const float* __restrict__ u, const float* __restrict__ Dv,
                      float* __restrict__ out) {
    __shared__ float Xs[64 * 68];     // [m][k]
    __shared__ float Cs[128 * 68];    // [h][k]
    const int tid  = threadIdx.x;
    const int lane = tid & 31;
    const int wave = tid >> 5;
    const int wm = wave & 3, wn = wave >> 2;
    const int r    = lane & 15;
    const int koff = (lane >> 4) << 1;
    const int rowBase = blockIdx.x * 64;
    const int r0 = tid >> 2, q = tid & 3;
    const int h0 = tid >> 1, q2 = tid & 1;

    v8f acc[4] = {};
    for (int kc = 0; kc < N2; kc += 64) {
        #pragma unroll
        for (int i = 0; i < 4; ++i) {
            int c4 = q + i * 4;
            float4 v = *(const float4*)(X2 + (size_t)(rowBase + r0) * N2 + kc + c4 * 4);
            *(float4*)(Xs + r0 * 68 + c4 * 4) = v;
        }
        #pragma unroll
        for (int i = 0; i < 8; ++i) {
            int c4 = q2 + i * 2;                  // 0..15
            float4 v = *(const float4*)(wct + (size_t)h0 * N2 + kc + c4 * 4);
            *(float4*)(Cs + h0 * 68 + c4 * 4) = v;
        }
        __syncthreads();
        #pragma unroll
        for (int k = 0; k < 64; k += 4) {
            int kk = k + koff;
            v2f a = *(const v2f*)(Xs + (wm * 16 + r) * 68 + kk);
            #pragma unroll
            for (int nt = 0; nt < 4; ++nt) {
                int col = wn * 64 + nt * 16 + r;
                v2f b = *(const v2f*)(Cs + col * 68 + kk);
                acc[nt] = __builtin_amdgcn_wmma_f32_16x16x4_f32(
                    false, a, false, b, (short)0, acc[nt], false, false);
            }
        }
        __syncthreads();
    }
    const int mhi = (lane >> 4) * 8;
    #pragma unroll
    for (int nt = 0; nt < 4; ++nt) {
        int col = wn * 64 + nt * 16 + r;        // 0..127 == h
        float dv = Dv[col];
        #pragma unroll
        for (int e = 0; e < 8; ++e) {
            int row = rowBase + wm * 16 + e + mhi;
            float skip = u[(size_t)row * H_DIM + col] * dv;
            out[(size_t)row * H_DIM + col] = acc[nt][e] + skip;
        }
    }
}

extern "C" void kernel_launch(void* const* d_in, const int* in_sizes, int n_in,
                              void* d_out, int out_size, void* d_ws, size_t ws_size,
                              hipStream_t stream) {
    const float* u      = (const float*)d_in[0];
    const float* A_diag = (const float*)d_in[1];
    const float* B      = (const float*)d_in[2];
    const float* C      = (const float*)d_in[3];
    const float* Dv     = (const float*)d_in[4];
    const float* steps  = (const float*)d_in[5];
    float* out = (float*)d_out;
    float* ws  = (float*)d_ws;

    prep_params<<<1, 256, 0, stream>>>(A_diag, steps, ws);
    build_w1<<<(N2 * H_DIM) / 256, 256, 0, stream>>>(B, ws);
    build_wc<<<(H_DIM * N2) / 256, 256, 0, stream>>>(C, ws);

    gemm1<<<dim3(L_SEQ / 64, N2 / 64), 256, 0, stream>>>(u, ws + OFF_W1, ws, ws + OFF_F1);

    scan_partial<<<NCHUNK, 512, 0, stream>>>(ws + OFF_F1, ws, ws + OFF_S);
    scan_carry<<<1, 512, 0, stream>>>(ws, ws + OFF_S, ws + OFF_CA);
    scan_final<<<NCHUNK, 512, 0, stream>>>(ws + OFF_F1, ws, ws + OFF_CA);

    gemm2<<<L_SEQ / 64, 256, 0, stream>>>(ws + OFF_F1, ws + OFF_WC, u, Dv, out);
}